// RegionProposalNetwork_25881472925951
// MI455X (gfx1250) — compile-verified
//
#include <hip/hip_runtime.h>
#include <stdint.h>

// ---------------- static config (matches reference formulas) ----------------
#define B_IMGS   4
#define NLVL     5
#define A_TOTAL  242991          // sum(ceil(800/s)*ceil(1216/s)*3)
#define KSEL     4741            // 1000+1000+1000+1000+741
#define NPAD     8192            // power-of-two pad for bitonic sort
#define NW       149             // ceil(KSEL/32) mask words per row
#define POST_NMS 1000
#define TIE_CAP  1024
#define CHUNK    8192            // elements per streaming block (32 KB LDS via TDM)
#define MAXCH    23              // ceil(182400 / CHUNK)
#define NEGV     (-1e10f)
#define NMS_TH   0.7f
#define W_IMGF   1216.0f
#define H_IMGF   800.0f
#define BBOX_CLIP 4.135166556742356f   // log(1000/16)

__constant__ int d_LVL_CNT[5]  = {182400, 45600, 11400, 2850, 741};
__constant__ int d_LVL_OFF[5]  = {0, 182400, 228000, 239400, 242250};
__constant__ int d_LVL_K[5]    = {1000, 1000, 1000, 1000, 741};
__constant__ int d_LVL_KOFF[5] = {0, 1000, 2000, 3000, 4000};

#ifndef __has_builtin
#define __has_builtin(x) 0
#endif
#if __has_builtin(__builtin_amdgcn_tensor_load_to_lds) && __has_builtin(__builtin_amdgcn_s_wait_tensorcnt)
#define USE_TDM 1
#else
#define USE_TDM 0
#endif

typedef __attribute__((ext_vector_type(4))) unsigned int tdm_uv4;
typedef __attribute__((ext_vector_type(8))) int          tdm_iv8;
typedef __attribute__((ext_vector_type(4))) int          tdm_iv4;

// Monotonic uint key: larger float -> larger key
__device__ __forceinline__ unsigned int fkey(float x) {
    unsigned int u = __float_as_uint(x);
    return u ^ ((unsigned int)((int)u >> 31) | 0x80000000u);
}

#if USE_TDM
// 1-D TDM copy: tensor_dwords of 4B in global; tile of tile_dwords into LDS.
// D# layout per CDNA5 ISA ch.8. OOB reads (tile > tensor) return zero.
// 6-arg builtin form (clang-23 / therock-10.0): g0, g1, g2, g3, g4, cpol.
__device__ __forceinline__ void tdm_load_dwords(const void* gptr, void* lptr,
                                                unsigned int tensor_dwords,
                                                unsigned int tile_dwords) {
    unsigned long long ga = (unsigned long long)(uintptr_t)gptr;
    unsigned int lds = (unsigned int)(uintptr_t)lptr;
    tdm_uv4 g0;
    g0[0] = 1u;                                        // count=1, user-mode, gather off
    g0[1] = lds;                                       // lds_addr
    g0[2] = (unsigned int)ga;                          // global_addr[31:0]
    g0[3] = (unsigned int)((ga >> 32) & 0x01FFFFFFull) // global_addr[56:32]
            | (2u << 30);                              // type=2 ("image")
    tdm_iv8 g1;
    g1[0] = (int)(2u << 16);                              // data_size=4B; wg_mask=0
    g1[1] = (int)((tensor_dwords & 0xFFFFu) << 16);       // tensor_dim0 lo16
    g1[2] = (int)((tensor_dwords >> 16) | (1u << 16));    // tensor_dim0 hi16 | tensor_dim1=1
    g1[3] = (int)((tile_dwords & 0xFFFFu) << 16);         // tensor_dim1 hi=0 | tile_dim0
    g1[4] = 0;                                            // tile_dim1=0 (1-D), tile_dim2=0
    g1[5] = (int)tensor_dwords;                           // tensor_dim0_stride lo32
    g1[6] = 0;
    g1[7] = 0;
    tdm_iv4 z4; z4[0]=0; z4[1]=0; z4[2]=0; z4[3]=0;
    tdm_iv8 z8; z8[0]=0; z8[1]=0; z8[2]=0; z8[3]=0; z8[4]=0; z8[5]=0; z8[6]=0; z8[7]=0;
    __builtin_amdgcn_tensor_load_to_lds(g0, g1, z4, z4, z8, 0);
}
#endif

// -------- kernel 1: reset select state + zero 256-bin global histograms ----
__global__ void rpn_init_kernel(unsigned int* prefix, int* krem, int* selCnt,
                                int* tieCnt, unsigned int* ghist) {
    int g = blockIdx.x, t = threadIdx.x;
    ghist[g * 256 + t] = 0u;
    if (g == 0 && t < B_IMGS * NLVL) {
        prefix[t] = 0u;
        krem[t]   = d_LVL_K[t % NLVL];
        selCnt[t] = 0;
        tieCnt[t] = 0;
    }
}

// -------- kernel 2a: chunked histogram pass (TDM-staged through LDS) -------
__global__ void rpn_radix_hist_kernel(const float* __restrict__ obj,
                                      const unsigned int* __restrict__ prefix,
                                      unsigned int* __restrict__ ghist, int shift) {
    __shared__ unsigned int hist[256];
    int g = blockIdx.y, b = g / NLVL, l = g % NLVL, t = threadIdx.x;
    int n = d_LVL_CNT[l];
    int start = blockIdx.x * CHUNK;
    if (start >= n) return;
    int cnt = min(CHUNK, n - start);
    const float* basep = obj + (size_t)b * A_TOTAL + d_LVL_OFF[l] + start;
    hist[t] = 0u;
#if USE_TDM
    __shared__ float buf[CHUNK];          // 32 KB staged by the tensor data mover
    if (t < 32) {                         // wave 0 issues the DMA once per block
        tdm_load_dwords(basep, buf, (unsigned int)cnt, CHUNK);
        __builtin_amdgcn_s_wait_tensorcnt(0);
    }
#endif
    __syncthreads();
    unsigned int pre    = prefix[g];
    unsigned int himask = (shift == 24) ? 0u : (0xFFFFFFFFu << (shift + 8));
    for (int i = t; i < cnt; i += 256) {
#if USE_TDM
        unsigned int k = fkey(buf[i]);
#else
        __builtin_prefetch(basep + i + 4096, 0, 0);   // global_prefetch_b8
        unsigned int k = fkey(basep[i]);
#endif
        if ((k & himask) == (pre & himask))
            atomicAdd(&hist[(k >> shift) & 255], 1u);
    }
    __syncthreads();
    if (hist[t]) atomicAdd(&ghist[g * 256 + t], hist[t]);
}

// -------- kernel 2b: pick digit, advance prefix/krem, re-zero bins ---------
__global__ void rpn_radix_pick_kernel(unsigned int* prefix, int* krem,
                                      unsigned int* ghist, int shift) {
    __shared__ unsigned int h[256];
    int g = blockIdx.x, t = threadIdx.x;
    h[t] = ghist[g * 256 + t];
    __syncthreads();
    if (t == 0) {
        int kr = krem[g], acc = 0, sel = 0;
        for (int d = 255; d >= 0; --d) { acc += (int)h[d]; if (acc >= kr) { sel = d; break; } }
        krem[g]   = kr - (acc - (int)h[sel]);
        prefix[g] = prefix[g] | ((unsigned int)sel << shift);
    }
    ghist[g * 256 + t] = 0u;              // ready for next pass
}

// -------- kernel 3: compact strict-greater + collect ties (TDM-staged) -----
__global__ void rpn_compact_kernel(const float* __restrict__ obj,
                                   const unsigned int* __restrict__ prefix,
                                   int* selCnt, int* tieCnt, int* tieBuf, int* selIdx) {
    int g = blockIdx.y, b = g / NLVL, l = g % NLVL, t = threadIdx.x;
    int n = d_LVL_CNT[l];
    int start = blockIdx.x * CHUNK;
    if (start >= n) return;
    int cnt = min(CHUNK, n - start);
    const float* basep = obj + (size_t)b * A_TOTAL + d_LVL_OFF[l] + start;
#if USE_TDM
    __shared__ float buf[CHUNK];
    if (t < 32) {
        tdm_load_dwords(basep, buf, (unsigned int)cnt, CHUNK);
        __builtin_amdgcn_s_wait_tensorcnt(0);
    }
    __syncthreads();
#endif
    unsigned int T = prefix[g];           // key of the k-th largest element
    int sbase = b * KSEL + d_LVL_KOFF[l];
    for (int i = t; i < cnt; i += 256) {
#if USE_TDM
        unsigned int k = fkey(buf[i]);
#else
        unsigned int k = fkey(basep[i]);
#endif
        if (k > T) {
            int p = atomicAdd(&selCnt[g], 1);
            selIdx[sbase + p] = d_LVL_OFF[l] + start + i;
        } else if (k == T) {
            int p = atomicAdd(&tieCnt[g], 1);
            if (p < TIE_CAP) tieBuf[g * TIE_CAP + p] = d_LVL_OFF[l] + start + i;
        }
    }
}

// -------- kernel 4: deterministic tie resolution (lowest index wins) -------
__global__ void rpn_resolve_kernel(const int* selCnt, const int* tieCnt,
                                   int* tieBuf, int* selIdx) {
    if (threadIdx.x != 0) return;
    int g = blockIdx.x, b = g / NLVL, l = g % NLVL;
    int have = selCnt[g];
    int need = d_LVL_K[l] - have;
    int tc   = tieCnt[g]; if (tc > TIE_CAP) tc = TIE_CAP;
    int* tb  = tieBuf + g * TIE_CAP;
    int base = b * KSEL + d_LVL_KOFF[l] + have;
    for (int m = 0; m < need; m++) {
        int best = 0x7FFFFFFF, bj = -1;
        for (int j = 0; j < tc; j++) { int v = tb[j]; if (v < best) { best = v; bj = j; } }
        if (bj >= 0) { tb[bj] = 0x7FFFFFFF; selIdx[base + m] = best; }
        else         { selIdx[base + m] = d_LVL_OFF[l]; }   // degenerate (tie overflow)
    }
}

// -------- kernel 5: decode+clip selected boxes, build sort keys ------------
__global__ void rpn_decode_kernel(const float* __restrict__ obj,
                                  const float* __restrict__ deltas,
                                  const float* __restrict__ anchors,
                                  const int* __restrict__ selIdx,
                                  float4* __restrict__ boxesRaw,
                                  float* __restrict__ scoreRaw,
                                  int* __restrict__ lvlRaw,
                                  unsigned int* __restrict__ skHi,
                                  unsigned short* __restrict__ skLo) {
    int gi = blockIdx.x * blockDim.x + threadIdx.x;
    if (gi >= B_IMGS * NPAD) return;
    int b = gi / NPAD, s = gi % NPAD;
    if (s >= KSEL) { skHi[gi] = 0xFFFFFFFFu; skLo[gi] = (unsigned short)s; return; }
    int lvl = (s >= 4000) ? 4 : (s / 1000);
    int idx = selIdx[b * KSEL + s];
    float sc = obj[(size_t)b * A_TOTAL + idx];
    const float4 a  = ((const float4*)anchors)[idx];
    const float4 dl = ((const float4*)deltas)[(size_t)b * A_TOTAL + idx];
    float wa = a.z - a.x, ha = a.w - a.y;
    float cxa = a.x + 0.5f * wa, cya = a.y + 0.5f * ha;
    float dw = fminf(dl.z, BBOX_CLIP), dh = fminf(dl.w, BBOX_CLIP);
    float cx = dl.x * wa + cxa, cy = dl.y * ha + cya;
    float w = expf(dw) * wa, h = expf(dh) * ha;
    float x1 = fminf(fmaxf(cx - 0.5f * w, 0.f), W_IMGF);
    float y1 = fminf(fmaxf(cy - 0.5f * h, 0.f), H_IMGF);
    float x2 = fminf(fmaxf(cx + 0.5f * w, 0.f), W_IMGF);
    float y2 = fminf(fmaxf(cy + 0.5f * h, 0.f), H_IMGF);
    bool valid = ((x2 - x1) >= 1e-3f) && ((y2 - y1) >= 1e-3f);
    float sm = valid ? sc : NEGV;
    boxesRaw[(size_t)b * KSEL + s] = make_float4(x1, y1, x2, y2);
    scoreRaw[(size_t)b * KSEL + s] = sm;
    lvlRaw  [(size_t)b * KSEL + s] = lvl;
    skHi[gi] = ~fkey(sm);                 // ascending sort => highest score first
    skLo[gi] = (unsigned short)s;         // stable tie-break by slot
}

// -------- kernel 6: per-batch bitonic sort (48KB LDS, TDM-staged) ----------
__global__ __launch_bounds__(1024)
void rpn_sort_kernel(const unsigned int* __restrict__ skHi,
                     const unsigned short* __restrict__ skLo,
                     const float4* __restrict__ boxesRaw,
                     const float* __restrict__ scoreRaw,
                     const int* __restrict__ lvlRaw,
                     float4* __restrict__ sBox, float* __restrict__ sScore,
                     int* __restrict__ sLvl) {
    __shared__ unsigned int   hi[NPAD];   // 32 KB
    __shared__ unsigned short lo[NPAD];   // 16 KB
    int b = blockIdx.x, t = threadIdx.x;
#if USE_TDM
    if (t < 32) {                         // two DMA transfers stage all 48 KB of keys
        tdm_load_dwords(skHi + (size_t)b * NPAD, hi, NPAD, NPAD);
        tdm_load_dwords(skLo + (size_t)b * NPAD, lo, NPAD / 2, NPAD / 2);
        __builtin_amdgcn_s_wait_tensorcnt(0);
    }
#else
    for (int i = t; i < NPAD; i += 1024) { hi[i] = skHi[b * NPAD + i]; lo[i] = skLo[b * NPAD + i]; }
#endif
    __syncthreads();
    for (int k = 2; k <= NPAD; k <<= 1) {
        for (int j = k >> 1; j > 0; j >>= 1) {
            for (int i = t; i < NPAD; i += 1024) {
                int ixj = i ^ j;
                if (ixj > i) {
                    bool up = ((i & k) == 0);
                    unsigned int   hA = hi[i], hB = hi[ixj];
                    unsigned short lA = lo[i], lB = lo[ixj];
                    bool gt = (hA > hB) || (hA == hB && lA > lB);
                    if (gt == up) { hi[i] = hB; hi[ixj] = hA; lo[i] = lB; lo[ixj] = lA; }
                }
            }
            __syncthreads();
        }
    }
    for (int i = t; i < KSEL; i += 1024) {
        int slot = lo[i];
        sBox  [(size_t)b * KSEL + i] = boxesRaw[(size_t)b * KSEL + slot];
        sScore[(size_t)b * KSEL + i] = scoreRaw[(size_t)b * KSEL + slot];
        sLvl  [(size_t)b * KSEL + i] = lvlRaw  [(size_t)b * KSEL + slot];
    }
}

// -------- kernel 7: NMS suppression bitmask (TDM column tiles) -------------
__global__ void rpn_nms_mask_kernel(const float4* __restrict__ sBox,
                                    const int* __restrict__ sLvl,
                                    unsigned int* __restrict__ maskBuf) {
    __shared__ float colBox[32 * 4];
    __shared__ int   colLvl[32];
    const int b  = blockIdx.z;
    const int cw = blockIdx.y;
    const int rt = blockIdx.x;
    const int c0 = cw * 32;
    const int t  = threadIdx.x;
    if (t < 32) {           // wave 0 only (wave32): one TDM issue per block
#if USE_TDM
        const float* gsrc = (const float*)(sBox + (size_t)b * KSEL + c0);
        unsigned int remDw = (unsigned int)(KSEL - c0) * 4u;   // OOB reads return 0
        tdm_load_dwords(gsrc, (void*)colBox, remDw, 128u);
#else
        int c = c0 + t;
        float4 v = (c < KSEL) ? sBox[(size_t)b * KSEL + c] : make_float4(0, 0, 0, 0);
        colBox[t * 4 + 0] = v.x; colBox[t * 4 + 1] = v.y;
        colBox[t * 4 + 2] = v.z; colBox[t * 4 + 3] = v.w;
#endif
        int c2 = c0 + t;
        colLvl[t] = (c2 < KSEL) ? sLvl[(size_t)b * KSEL + c2] : -1;
#if USE_TDM
        __builtin_amdgcn_s_wait_tensorcnt(0);
#endif
    }
    __syncthreads();
    int r = rt * 64 + t;
    if (r >= KSEL) return;
    float4 rb = sBox[(size_t)b * KSEL + r];
    int    rl = sLvl[(size_t)b * KSEL + r];
    float rArea = (rb.z - rb.x) * (rb.w - rb.y);
    unsigned int word = 0u;
    for (int j = 0; j < 32; j++) {
        int c = c0 + j;
        if (c <= r || c >= KSEL) continue;
        if (colLvl[j] != rl) continue;     // level-offset trick == same-level test
        float cx1 = colBox[j * 4 + 0], cy1 = colBox[j * 4 + 1];
        float cx2 = colBox[j * 4 + 2], cy2 = colBox[j * 4 + 3];
        float ix1 = fmaxf(rb.x, cx1), iy1 = fmaxf(rb.y, cy1);
        float ix2 = fminf(rb.z, cx2), iy2 = fminf(rb.w, cy2);
        float iw = fmaxf(ix2 - ix1, 0.f), ih = fmaxf(iy2 - iy1, 0.f);
        float inter = iw * ih;
        float cArea = (cx2 - cx1) * (cy2 - cy1);
        float iou = inter / (rArea + cArea - inter);
        if (iou > NMS_TH) word |= (1u << j);
    }
    maskBuf[((size_t)b * KSEL + r) * NW + cw] = word;
}

// -------- kernel 8: sequential keep-propagation + output -------------------
__global__ void rpn_nms_reduce_kernel(const float4* __restrict__ sBox,
                                      const float* __restrict__ sScore,
                                      const unsigned int* __restrict__ maskBuf,
                                      float* __restrict__ out) {
    __shared__ unsigned int remv[NW];
    __shared__ int sIdx[POST_NMS];
    __shared__ int sCnt;
    __shared__ int sKeep;
    int b = blockIdx.x, t = threadIdx.x;
    for (int i = t; i < NW; i += blockDim.x) remv[i] = 0u;
    if (t == 0) sCnt = 0;
    __syncthreads();
    for (int i = 0; i < KSEL; i++) {
        if (t == 0) {
            bool sup   = (remv[i >> 5] >> (i & 31)) & 1u;
            bool valid = sScore[(size_t)b * KSEL + i] > -1e9f;
            int  kp    = (valid && !sup) ? 1 : 0;
            sKeep = kp;
            if (kp) { if (sCnt < POST_NMS) sIdx[sCnt] = i; sCnt++; }
        }
        __syncthreads();
        if (sKeep) {
            const unsigned int* row = maskBuf + ((size_t)b * KSEL + i) * NW;
            for (int w = t; w < NW; w += blockDim.x) remv[w] |= row[w];
        }
        __syncthreads();
    }
    int cnt = sCnt; if (cnt > POST_NMS) cnt = POST_NMS;
    float* outB = out;                                 // [B,1000,4]
    float* outS = out + (size_t)B_IMGS * POST_NMS * 4; // [B,1000]
    for (int o = t; o < POST_NMS; o += blockDim.x) {
        size_t ob = ((size_t)b * POST_NMS + o) * 4;
        if (o < cnt) {
            int i = sIdx[o];
            float4 bx = sBox[(size_t)b * KSEL + i];
            outB[ob + 0] = bx.x; outB[ob + 1] = bx.y;
            outB[ob + 2] = bx.z; outB[ob + 3] = bx.w;
            outS[(size_t)b * POST_NMS + o] = sScore[(size_t)b * KSEL + i];
        } else {
            outB[ob + 0] = 0.f; outB[ob + 1] = 0.f; outB[ob + 2] = 0.f; outB[ob + 3] = 0.f;
            outS[(size_t)b * POST_NMS + o] = NEGV;
        }
    }
}

// ---------------------------------------------------------------------------
extern "C" void kernel_launch(void* const* d_in, const int* in_sizes, int n_in,
                              void* d_out, int out_size, void* d_ws, size_t ws_size,
                              hipStream_t stream) {
    (void)in_sizes; (void)n_in; (void)out_size; (void)ws_size;
    const float* obj     = (const float*)d_in[0];
    const float* deltas  = (const float*)d_in[1];
    const float* anchors = (const float*)d_in[2];

    size_t off = 0;
    auto arena = [&](size_t bytes) -> void* {
        off = (off + 255) & ~(size_t)255;
        void* p = (char*)d_ws + off;
        off += bytes;
        return p;
    };
    unsigned int*   prefix   = (unsigned int*)  arena(20 * sizeof(unsigned int));
    int*            krem     = (int*)           arena(20 * sizeof(int));
    int*            selCnt   = (int*)           arena(20 * sizeof(int));
    int*            tieCnt   = (int*)           arena(20 * sizeof(int));
    unsigned int*   ghist    = (unsigned int*)  arena((size_t)20 * 256 * sizeof(unsigned int));
    int*            tieBuf   = (int*)           arena((size_t)20 * TIE_CAP * sizeof(int));
    int*            selIdx   = (int*)           arena((size_t)B_IMGS * KSEL * sizeof(int));
    float4*         boxesRaw = (float4*)        arena((size_t)B_IMGS * KSEL * sizeof(float4));
    float*          scoreRaw = (float*)         arena((size_t)B_IMGS * KSEL * sizeof(float));
    int*            lvlRaw   = (int*)           arena((size_t)B_IMGS * KSEL * sizeof(int));
    unsigned int*   skHi     = (unsigned int*)  arena((size_t)B_IMGS * NPAD * sizeof(unsigned int));
    unsigned short* skLo     = (unsigned short*)arena((size_t)B_IMGS * NPAD * sizeof(unsigned short));
    float4*         sBox     = (float4*)        arena((size_t)B_IMGS * KSEL * sizeof(float4));
    float*          sScore   = (float*)         arena((size_t)B_IMGS * KSEL * sizeof(float));
    int*            sLvl     = (int*)           arena((size_t)B_IMGS * KSEL * sizeof(int));
    unsigned int*   maskBuf  = (unsigned int*)  arena((size_t)B_IMGS * KSEL * NW * sizeof(unsigned int));

    rpn_init_kernel<<<B_IMGS * NLVL, 256, 0, stream>>>(prefix, krem, selCnt, tieCnt, ghist);
    dim3 cg(MAXCH, B_IMGS * NLVL);
    for (int shift = 24; shift >= 0; shift -= 8) {
        rpn_radix_hist_kernel<<<cg, 256, 0, stream>>>(obj, prefix, ghist, shift);
        rpn_radix_pick_kernel<<<B_IMGS * NLVL, 256, 0, stream>>>(prefix, krem, ghist, shift);
    }
    rpn_compact_kernel<<<cg, 256, 0, stream>>>(obj, prefix, selCnt, tieCnt, tieBuf, selIdx);
    rpn_resolve_kernel<<<B_IMGS * NLVL, 64, 0, stream>>>(selCnt, tieCnt, tieBuf, selIdx);
    rpn_decode_kernel<<<(B_IMGS * NPAD + 255) / 256, 256, 0, stream>>>(
        obj, deltas, anchors, selIdx, boxesRaw, scoreRaw, lvlRaw, skHi, skLo);
    rpn_sort_kernel<<<B_IMGS, 1024, 0, stream>>>(skHi, skLo, boxesRaw, scoreRaw, lvlRaw,
                                                 sBox, sScore, sLvl);
    dim3 mg((KSEL + 63) / 64, NW, B_IMGS);
    rpn_nms_mask_kernel<<<mg, 64, 0, stream>>>(sBox, sLvl, maskBuf);
    rpn_nms_reduce_kernel<<<B_IMGS, 256, 0, stream>>>(sBox, sScore, maskBuf, (float*)d_out);
}